// GraphConvolution_P2_31250182046301
// MI455X (gfx1250) — compile-verified
//
#include <hip/hip_runtime.h>

#define D_FEAT 128
#define CHUNK 512
#define EDGES_PER_BLOCK 2048
#define NWAVES 8

typedef unsigned int u32x4 __attribute__((ext_vector_type(4)));
typedef int          i32x8 __attribute__((ext_vector_type(8)));
typedef int          i32x4 __attribute__((ext_vector_type(4)));

// Raw LDS byte offset: low 32 bits of a flat LDS pointer are the offset within
// the workgroup's LDS allocation (ISA 10.2 aperture rules).
__device__ __forceinline__ unsigned lds_offset(const void* p) {
  return (unsigned)(unsigned long long)(uintptr_t)p;
}

// Issue one TDM 1-D copy of `nelem` 4-byte elements: global -> LDS.
// D# packing per CDNA5 ISA 8.3/8.4 (group0 128b, group1 256b, groups 2/3 unused).
__device__ __forceinline__ void tdm_load_1d(const void* gsrc, unsigned lds_addr,
                                            unsigned nelem) {
  unsigned long long ga = (unsigned long long)(uintptr_t)gsrc;
  u32x4 g0;
  g0.x = 1u;                                                  // count=1, user D#
  g0.y = lds_addr;                                            // lds_addr [63:32]
  g0.z = (unsigned)ga;                                        // global_addr[31:0]
  g0.w = ((unsigned)(ga >> 32) & 0x01FFFFFFu) | (2u << 30);   // addr[56:32] | type=2
  i32x8 g1;
  g1[0] = (2 << 16);                                          // data_size = 4 bytes
  g1[1] = (int)((nelem & 0xFFFFu) << 16);                     // tensor_dim0 lo16 @[63:48]
  g1[2] = (int)(((nelem >> 16) & 0xFFFFu) | (1u << 16));      // tensor_dim0 hi16, tensor_dim1=1
  g1[3] = (int)((nelem & 0xFFFFu) << 16);                     // tile_dim0 = nelem @[127:112]
  g1[4] = 1;                                                  // tile_dim1 = 1, tile_dim2 = 0
  g1[5] = (int)nelem;                                         // tensor_dim0_stride lo32
  g1[6] = 0;
  g1[7] = 0;
  i32x4 z4 = (i32x4)0;                                        // groups 2/3 unused (<=2D)
  i32x8 z8 = (i32x8)0;                                        // 5th descriptor arg (clang-23 form)
  __builtin_amdgcn_tensor_load_to_lds(g0, g1, z4, z4, z8, 0);
}

__global__ void __launch_bounds__(256) gc_zero_kernel(float4* __restrict__ p,
                                                      long long n4) {
  long long i = (long long)blockIdx.x * blockDim.x + threadIdx.x;
  if (i < n4) p[i] = make_float4(0.f, 0.f, 0.f, 0.f);
}

__global__ void __launch_bounds__(256) gc_spmm_kernel(
    const float* __restrict__ support,   // [N, 128]
    const float* __restrict__ vals,      // [E]
    const int*   __restrict__ rows,      // [E]
    const int*   __restrict__ cols,      // [E]
    float*       __restrict__ out,       // [N, 128], pre-zeroed
    int n_edges) {
  __shared__ int   s_rows[2][CHUNK];
  __shared__ int   s_cols[2][CHUNK];
  __shared__ float s_vals[2][CHUNK];

  const int tid  = threadIdx.x;
  const int lane = tid & 31;
  const int wave = __builtin_amdgcn_readfirstlane(tid >> 5);  // wave-uniform SGPR

  const long long blockBase = (long long)blockIdx.x * EDGES_PER_BLOCK;
  long long remll = (long long)n_edges - blockBase;
  const int blockEdges = remll > EDGES_PER_BLOCK ? EDGES_PER_BLOCK : (int)remll;
  const int nchunks = (blockEdges + CHUNK - 1) / CHUNK;

  // Prologue: wave 0 primes the two-deep TDM pipeline (EXEC-independent DMA,
  // issued from a scalar-branched wave so exactly one DMA set per chunk).
  if (wave == 0) {
    const int pre = nchunks < 2 ? nchunks : 2;
    for (int c = 0; c < pre; ++c) {
      const int base = c * CHUNK;
      const unsigned n = (unsigned)min(CHUNK, blockEdges - base);
      tdm_load_1d(rows + blockBase + base, lds_offset(&s_rows[c][0]), n);
      tdm_load_1d(cols + blockBase + base, lds_offset(&s_cols[c][0]), n);
      tdm_load_1d(vals + blockBase + base, lds_offset(&s_vals[c][0]), n);
    }
  }

  for (int c = 0; c < nchunks; ++c) {
    const int buf = c & 1;
    if (wave == 0) {
      // 3 TDM ops per chunk, in-order per wave: chunk c done when <= 3 pending
      // (the prefetched chunk c+1), or 0 when nothing further was issued.
      if (c + 1 < nchunks) __builtin_amdgcn_s_wait_tensorcnt(3);
      else                 __builtin_amdgcn_s_wait_tensorcnt(0);
    }
    __syncthreads();

    const int base = c * CHUNK;
    const int n = min(CHUNK, blockEdges - base);
    // One edge per wave-iteration: 32 lanes x float4 cover all 128 features.
    // Gather is one coalesced 512B line; scatter is 4 coalesced f32 atomics/lane.
#pragma unroll 2
    for (int j = wave; j < n; j += NWAVES) {
      const int   r  = s_rows[buf][j];
      const int   ci = s_cols[buf][j];
      const float v  = s_vals[buf][j];
      const float4 f = ((const float4*)(support + (long long)ci * D_FEAT))[lane];
      float* op = out + (long long)r * D_FEAT + lane * 4;
      atomicAdd(op + 0, f.x * v);
      atomicAdd(op + 1, f.y * v);
      atomicAdd(op + 2, f.z * v);
      atomicAdd(op + 3, f.w * v);
    }
    __syncthreads();

    // Refill the buffer we just drained with chunk c+2.
    if (wave == 0 && c + 2 < nchunks) {
      const int base2 = (c + 2) * CHUNK;
      const unsigned n2 = (unsigned)min(CHUNK, blockEdges - base2);
      tdm_load_1d(rows + blockBase + base2, lds_offset(&s_rows[buf][0]), n2);
      tdm_load_1d(cols + blockBase + base2, lds_offset(&s_cols[buf][0]), n2);
      tdm_load_1d(vals + blockBase + base2, lds_offset(&s_vals[buf][0]), n2);
    }
  }
}

extern "C" void kernel_launch(void* const* d_in, const int* in_sizes, int n_in,
                              void* d_out, int out_size, void* d_ws, size_t ws_size,
                              hipStream_t stream) {
  (void)n_in; (void)d_ws; (void)ws_size;
  const float* support = (const float*)d_in[0];
  const float* vals    = (const float*)d_in[1];
  const int*   rows    = (const int*)d_in[2];
  const int*   cols    = (const int*)d_in[3];
  float*       out     = (float*)d_out;
  const int n_edges    = in_sizes[1];

  // Zero the (poisoned) output before scatter-add.
  const long long n4 = (long long)out_size / 4;
  const int zblocks = (int)((n4 + 255) / 256);
  gc_zero_kernel<<<zblocks, 256, 0, stream>>>((float4*)out, n4);

  const int nblocks = (n_edges + EDGES_PER_BLOCK - 1) / EDGES_PER_BLOCK;
  gc_spmm_kernel<<<nblocks, 256, 0, stream>>>(support, vals, rows, cols, out,
                                              n_edges);
}